// AdaMix_54795192762733
// MI455X (gfx1250) — compile-verified
//
#include <hip/hip_runtime.h>

// ---------------------------------------------------------------------------
// AdaMix patch mixing for MI455X (gfx1250, wave32).
// Memory-bound (~320 MB traffic -> ~14us @ 23.3 TB/s). WMMA f32 16x16x4 is
// used as the wave32 cross-lane reduction primitive for the dice-loss class
// sums and the per-patch confidence means.
// ---------------------------------------------------------------------------

#define B_    64
#define C_    3
#define IMG_  256
#define NC_   4
#define NH_   8
#define P_    64
#define NPIX  (IMG_ * IMG_)   // 65536

typedef __attribute__((ext_vector_type(2))) float v2f;
typedef __attribute__((ext_vector_type(8))) float v8f;

// Sum of v over all 32 lanes of the wave, result broadcast to every lane.
// A[m][0] = v_m (lanes 0-15 VGPR0), A[m][2] = v_{m+16} (lanes 16-31 VGPR0),
// A[*][1] = A[*][3] = 0.  B = ones  =>  D[m][n] = v_m + v_{m+16}.
// Per-lane sum of the 8 D VGPRs gives half-sums; shfl_xor(16) completes it.
// Requires EXEC == all ones (callers use full 256-thread blocks, no
// divergence at the call site).
__device__ __forceinline__ float wave_sum32(float v) {
    v2f a;  a[0] = v;    a[1] = 0.0f;
    v2f bo; bo[0] = 1.0f; bo[1] = 1.0f;
    v8f c = {};
    v8f d = __builtin_amdgcn_wmma_f32_16x16x4_f32(
        /*neg_a=*/false, a, /*neg_b=*/false, bo,
        /*c_mod=*/(short)0, c, /*reuse_a=*/false, /*reuse_b=*/false);
    float w = ((d[0] + d[1]) + (d[2] + d[3])) + ((d[4] + d[5]) + (d[6] + d[7]));
    w += __shfl_xor(w, 16, 32);
    return w;
}

// ---------------------------------------------------------------------------
// Kernel 1: per-sample dice loss -> sign[b] (sort direction), k[b] (mix count)
// ---------------------------------------------------------------------------
__global__ __launch_bounds__(256) void dice_kernel(
        const float* __restrict__ pred,    // [B, NC, IMG, IMG]
        const int*   __restrict__ olabel,  // [B, IMG, IMG]
        float* __restrict__ signArr,       // [B]
        int*   __restrict__ kArr) {        // [B]
    const int b   = blockIdx.x;
    const int tid = threadIdx.x;
    const float* pb = pred   + (size_t)b * NC_ * NPIX;
    const int*   lb = olabel + (size_t)b * NPIX;

    float inter0 = 0.f, inter1 = 0.f, inter2 = 0.f, inter3 = 0.f;
    float uni0   = 0.f, uni1   = 0.f, uni2   = 0.f, uni3   = 0.f;

    for (int idx = tid; idx < NPIX; idx += 256) {
        float p0 = pb[idx];
        float p1 = pb[idx +     NPIX];
        float p2 = pb[idx + 2 * NPIX];
        float p3 = pb[idx + 3 * NPIX];
        int   l  = lb[idx];
        float m  = fmaxf(fmaxf(p0, p1), fmaxf(p2, p3));
        float e0 = __expf(p0 - m), e1 = __expf(p1 - m);
        float e2 = __expf(p2 - m), e3 = __expf(p3 - m);
        float inv = 1.0f / (e0 + e1 + e2 + e3);
        float s0 = e0 * inv, s1 = e1 * inv, s2 = e2 * inv, s3 = e3 * inv;
        float t0 = (l == 0) ? 1.0f : 0.0f;
        float t1 = (l == 1) ? 1.0f : 0.0f;
        float t2 = (l == 2) ? 1.0f : 0.0f;
        float t3 = (l == 3) ? 1.0f : 0.0f;
        uni0 += s0 + t0;  uni1 += s1 + t1;  uni2 += s2 + t2;  uni3 += s3 + t3;
        inter0 += s0 * t0; inter1 += s1 * t1; inter2 += s2 * t2; inter3 += s3 * t3;
    }

    // Wave-level WMMA reductions (EXEC is all ones: no divergence here).
    float vals[8] = {inter0, inter1, inter2, inter3, uni0, uni1, uni2, uni3};
    __shared__ float red[8][8];
    const int wave = tid >> 5;
    const int lane = tid & 31;
#pragma unroll
    for (int j = 0; j < 8; ++j) {
        float t = wave_sum32(vals[j]);
        if (lane == 0) red[wave][j] = t;
    }
    __syncthreads();

    if (tid == 0) {
        float dice = 0.0f;
#pragma unroll
        for (int cc = 0; cc < 4; ++cc) {
            float I = 0.f, U = 0.f;
#pragma unroll
            for (int w = 0; w < 8; ++w) { I += red[w][cc]; U += red[w][4 + cc]; }
            dice += 2.0f * I / (U + 1e-5f);
        }
        dice *= 0.25f;
        float loss = 1.0f - dice;
        float sg   = (loss < 1.0f) ? -1.0f : 1.0f;           // AGE = 1.0
        float sp_w = 1.0f - loss / 1.00001f;                 // AGE + 1e-5
        int   kk   = (int)fabsf(truncf(16.0f * sp_w));
        if (kk > 16) kk = 16;
        signArr[b] = sg;
        kArr[b]    = kk;
    }
}

// ---------------------------------------------------------------------------
// Kernel 2: per-patch confidence means (original & augmented)
// ---------------------------------------------------------------------------
__global__ __launch_bounds__(256) void mean_kernel(
        const float* __restrict__ oconf, const float* __restrict__ aconf,
        float* __restrict__ omean, float* __restrict__ amean) {
    const int bp = blockIdx.x;          // b * 64 + p
    const int b  = bp >> 6;
    const int p  = bp & 63;
    const int py = p >> 3, px = p & 7;
    const float* ob = oconf + (size_t)b * NPIX;
    const float* ab = aconf + (size_t)b * NPIX;
    const int tid = threadIdx.x;

    float so = 0.f, sa = 0.f;
#pragma unroll
    for (int idx = tid; idx < 1024; idx += 256) {
        int i = idx >> 5, j = idx & 31;
        int off = (py * 32 + i) * IMG_ + px * 32 + j;
        so += ob[off];
        sa += ab[off];
    }
    so = wave_sum32(so);
    sa = wave_sum32(sa);

    __shared__ float red[8][2];
    const int wave = tid >> 5, lane = tid & 31;
    if (lane == 0) { red[wave][0] = so; red[wave][1] = sa; }
    __syncthreads();
    if (tid == 0) {
        float to = 0.f, ta = 0.f;
#pragma unroll
        for (int w = 0; w < 8; ++w) { to += red[w][0]; ta += red[w][1]; }
        omean[bp] = to * (1.0f / 1024.0f);
        amean[bp] = ta * (1.0f / 1024.0f);
    }
}

// ---------------------------------------------------------------------------
// Kernel 3: stable rank sort (matches jnp.argsort stability) + source map.
// src[b][q] = q                (keep original patch q)
//           = 64 + s           (take augmented patch s)
// ---------------------------------------------------------------------------
__global__ __launch_bounds__(64) void rank_kernel(
        const float* __restrict__ omean, const float* __restrict__ amean,
        const float* __restrict__ signArr, const int* __restrict__ kArr,
        int* __restrict__ src) {
    const int b = blockIdx.x;
    const int t = threadIdx.x;
    __shared__ float ko[P_], ka[P_];
    __shared__ int aofr[P_];

    const float sg = signArr[b];
    ko[t] =  sg * omean[b * P_ + t];   // ascending sort of sign*omean
    ka[t] = -sg * amean[b * P_ + t];   // ascending sort of -sign*amean
    __syncthreads();

    const float mo = ko[t], ma = ka[t];
    int ro = 0, ra = 0;
#pragma unroll 4
    for (int j = 0; j < P_; ++j) {
        float vo = ko[j], va = ka[j];
        ro += (vo < mo) || (vo == mo && j < t);   // stable tie-break
        ra += (va < ma) || (va == ma && j < t);
    }
    aofr[ra] = t;          // inverse permutation: rank -> augmented patch
    __syncthreads();

    const int kk = kArr[b];
    src[b * P_ + t] = (ro < kk) ? (64 + aofr[ro]) : t;
}

// ---------------------------------------------------------------------------
// Kernel 4: assemble output. One thread per pixel, all 5 channels.
// Wave = one 32-pixel patch row -> uniform branch, coalesced 128B loads.
// ---------------------------------------------------------------------------
__global__ __launch_bounds__(256) void mix_kernel(
        const float* __restrict__ oimage, const float* __restrict__ aimage,
        const int*   __restrict__ olabel, const int*   __restrict__ alabel,
        const float* __restrict__ oconf,  const float* __restrict__ aconf,
        const int*   __restrict__ src,
        float* __restrict__ outImg, int* __restrict__ outLbl,
        float* __restrict__ outConf) {
    const int gid = blockIdx.x * 256 + threadIdx.x;
    const int b   = gid >> 16;          // NPIX = 65536
    const int rem = gid & (NPIX - 1);
    const int y   = rem >> 8;
    const int x   = rem & 255;
    const int q   = ((y >> 5) << 3) + (x >> 5);
    const int s   = src[b * P_ + q];

    const float* img; const int* lbl; const float* cf;
    int soff;
    if (s >= 64) {
        const int sp = s - 64;
        const int sy = ((sp >> 3) << 5) + (y & 31);
        const int sx = ((sp & 7)  << 5) + (x & 31);
        soff = sy * IMG_ + sx;
        img = aimage; lbl = alabel; cf = aconf;
    } else {
        soff = rem;
        img = oimage; lbl = olabel; cf = oconf;
    }

    const size_t chanBase = (size_t)b * C_ * NPIX;
#pragma unroll
    for (int cch = 0; cch < C_; ++cch)
        outImg[chanBase + (size_t)cch * NPIX + rem] =
            img[chanBase + (size_t)cch * NPIX + soff];
    outLbl[gid]  = lbl[(size_t)b * NPIX + soff];
    outConf[gid] = cf [(size_t)b * NPIX + soff];
}

// ---------------------------------------------------------------------------
extern "C" void kernel_launch(void* const* d_in, const int* in_sizes, int n_in,
                              void* d_out, int out_size, void* d_ws, size_t ws_size,
                              hipStream_t stream) {
    const float* oimage = (const float*)d_in[0];
    const float* aimage = (const float*)d_in[1];
    const int*   olabel = (const int*)  d_in[2];
    const int*   alabel = (const int*)  d_in[3];
    const float* oconf  = (const float*)d_in[4];
    const float* aconf  = (const float*)d_in[5];
    const float* pred   = (const float*)d_in[6];
    // d_in[7] = cur_step (unused by the computed branch)

    // Workspace layout (floats):
    float* ws      = (float*)d_ws;
    float* omean   = ws;                 // 4096
    float* amean   = ws + 4096;          // 4096
    float* signArr = ws + 8192;          // 64
    int*   kArr    = (int*)(ws + 8256);  // 64
    int*   src     = (int*)(ws + 8320);  // 4096

    // Output layout: [image f32 | label i32 bits | conf f32]
    float* outImg  = (float*)d_out;
    int*   outLbl  = (int*)((float*)d_out + (size_t)B_ * C_ * NPIX);
    float* outConf = (float*)d_out + (size_t)B_ * C_ * NPIX + (size_t)B_ * NPIX;

    dice_kernel<<<B_, 256, 0, stream>>>(pred, olabel, signArr, kArr);
    mean_kernel<<<B_ * P_, 256, 0, stream>>>(oconf, aconf, omean, amean);
    rank_kernel<<<B_, P_, 0, stream>>>(omean, amean, signArr, kArr, src);
    mix_kernel<<<(B_ * NPIX) / 256, 256, 0, stream>>>(
        oimage, aimage, olabel, alabel, oconf, aconf, src,
        outImg, outLbl, outConf);
}